// RegimeGatingNetwork_8486855376990
// MI455X (gfx1250) — compile-verified
//
#include <hip/hip_runtime.h>

typedef __attribute__((ext_vector_type(2))) float v2f;
typedef __attribute__((ext_vector_type(8))) float v8f;

#define B_TOTAL   16384
#define T_DIM     512
#define F_DIM     8
#define NEXP      4
#define ALPHA_F   (2.0f / 11.0f)
#define LN_EPS_F  1e-3f
// log2(9/11) = log2(1 - ALPHA)
#define L2OM      (-0.2895066187F)

// Block: 256 threads = 8 waves.
//   wave = 2 batch-tiles (16 rows each) x 4 T-chunks (128 t-steps each)
// Each wave accumulates a partial 16x16 WMMA D-tile whose columns 0/1 hold
// the partial EMA of channel 0 (hurst) / channel 1 (gk) for its 16 rows.
__global__ __launch_bounds__(256) void regime_gating_kernel(
    const float* __restrict__ x,
    const float* __restrict__ W1, const float* __restrict__ b1,
    const float* __restrict__ gamma, const float* __restrict__ beta,
    const float* __restrict__ W2, const float* __restrict__ b2,
    const float* __restrict__ W3, const float* __restrict__ b3,
    float* __restrict__ out)
{
    __shared__ float w_lds[T_DIM];          // EMA weights, absolute per t
    __shared__ float part[2][4][32][8];     // [tile][tchunk][lane][vgpr]
    __shared__ float regime[2][16][2];      // [tile][row][channel]

    const int tid = threadIdx.x;

    // ---- precompute EMA weights: w[t] = a*(1-a)^(511-t), w[0]=(1-a)^511 ----
    for (int t = tid; t < T_DIM; t += 256) {
        float e = exp2f((float)(T_DIM - 1 - t) * L2OM);
        w_lds[t] = (t == 0) ? e : ALPHA_F * e;
    }
    __syncthreads();

    const int lane   = tid & 31;
    const int wave   = tid >> 5;
    const int tile   = wave >> 2;    // 0..1 : which 16-row batch tile
    const int tchunk = wave & 3;     // 0..3 : which quarter of T
    const int col    = lane & 15;    // M row (A) / N column (B,D)
    const int half   = lane >> 4;    // 0: K=0,1   1: K=2,3

    const int   brow = blockIdx.x * 32 + tile * 16 + col;
    const float* xb  = x + (size_t)brow * (T_DIM * F_DIM);
    const int   tbase = tchunk * 128;

    // ---- WMMA reduction over this wave's 128 time steps (2 t per WMMA) ----
    // Unroll 8: keeps 8 independent global B64 loads in flight per wave for
    // latency hiding against the ~268 MB HBM stream this kernel is bound by.
    v8f acc = {0.f, 0.f, 0.f, 0.f, 0.f, 0.f, 0.f, 0.f};
    #pragma unroll 8
    for (int i = 0; i < 64; ++i) {
        const int t = tbase + 2 * i + half;
        // A: lane holds x[brow, t0+half, 0:2]  (K = 2*half + {0,1})
        float2 av = *reinterpret_cast<const float2*>(xb + (size_t)t * F_DIM);
        v2f a; a.x = av.x; a.y = av.y;
        // B: col 0 selects channel 0 (even K), col 1 selects channel 1 (odd K)
        const float wh = w_lds[t];
        v2f bm;
        bm.x = (col == 0) ? wh : 0.f;   // B[2*half+0, col]
        bm.y = (col == 1) ? wh : 0.f;   // B[2*half+1, col]
        acc = __builtin_amdgcn_wmma_f32_16x16x4_f32(
                  false, a, false, bm, (short)0, acc, false, false);
    }

    // ---- stash partial D-tile ----
    #pragma unroll
    for (int j = 0; j < 8; ++j) part[tile][tchunk][lane][j] = acc[j];
    __syncthreads();

    // ---- combine the 4 T-chunk partials into regime[tile][m][ch] ----
    // D layout: element (m, n) lives in VGPR (m & 7), lane n + 16*(m >= 8)
    if (tid < 64) {
        const int tl = tid >> 5;
        const int m  = (tid >> 1) & 15;
        const int ch = tid & 1;
        const int ln = ch + ((m >> 3) << 4);
        const int j  = m & 7;
        regime[tl][m][ch] = part[tl][0][ln][j] + part[tl][1][ln][j]
                          + part[tl][2][ln][j] + part[tl][3][ln][j];
    }
    __syncthreads();

    // ---- tiny MLP epilogue: one thread per batch row (32 rows/block) ----
    if (tid < 32) {
        const int tl = tid >> 4;
        const int m  = tid & 15;
        const int b  = blockIdx.x * 32 + tid;
        const float r0 = regime[tl][m][0];
        const float r1 = regime[tl][m][1];

        // layer 1: (2 -> 32) + relu
        float g[32];
        float mu = 0.f;
        #pragma unroll
        for (int j = 0; j < 32; ++j) {
            float v = fmaf(r0, W1[j], fmaf(r1, W1[32 + j], b1[j]));
            v = fmaxf(v, 0.f);
            g[j] = v;
            mu += v;
        }
        mu *= (1.0f / 32.0f);
        float var = 0.f;
        #pragma unroll
        for (int j = 0; j < 32; ++j) { float d = g[j] - mu; var += d * d; }
        var *= (1.0f / 32.0f);
        const float inv = rsqrtf(var + LN_EPS_F);
        #pragma unroll
        for (int j = 0; j < 32; ++j)
            g[j] = (g[j] - mu) * inv * gamma[j] + beta[j];

        // layer 2: (32 -> 16) + relu
        float h[16];
        #pragma unroll
        for (int k = 0; k < 16; ++k) {
            float s = b2[k];
            #pragma unroll
            for (int j = 0; j < 32; ++j) s = fmaf(g[j], W2[j * 16 + k], s);
            h[k] = fmaxf(s, 0.f);
        }

        // layer 3: (16 -> 4) + softmax
        float o[NEXP];
        #pragma unroll
        for (int e = 0; e < NEXP; ++e) {
            float s = b3[e];
            #pragma unroll
            for (int k = 0; k < 16; ++k) s = fmaf(h[k], W3[k * NEXP + e], s);
            o[e] = s;
        }
        float mx = fmaxf(fmaxf(o[0], o[1]), fmaxf(o[2], o[3]));
        float e0 = expf(o[0] - mx), e1 = expf(o[1] - mx);
        float e2 = expf(o[2] - mx), e3 = expf(o[3] - mx);
        const float invs = 1.0f / (e0 + e1 + e2 + e3);
        out[(size_t)b * NEXP + 0] = e0 * invs;
        out[(size_t)b * NEXP + 1] = e1 * invs;
        out[(size_t)b * NEXP + 2] = e2 * invs;
        out[(size_t)b * NEXP + 3] = e3 * invs;
    }
}

extern "C" void kernel_launch(void* const* d_in, const int* in_sizes, int n_in,
                              void* d_out, int out_size, void* d_ws, size_t ws_size,
                              hipStream_t stream) {
    (void)in_sizes; (void)n_in; (void)d_ws; (void)ws_size; (void)out_size;
    const float* x     = (const float*)d_in[0];
    const float* W1    = (const float*)d_in[1];
    const float* b1    = (const float*)d_in[2];
    const float* gamma = (const float*)d_in[3];
    const float* beta  = (const float*)d_in[4];
    const float* W2    = (const float*)d_in[5];
    const float* b2    = (const float*)d_in[6];
    const float* W3    = (const float*)d_in[7];
    const float* b3    = (const float*)d_in[8];
    float* out = (float*)d_out;

    dim3 grid(B_TOTAL / 32);   // 512 blocks, 32 batch rows per block
    dim3 block(256);           // 8 waves: 2 tiles x 4 T-chunks
    regime_gating_kernel<<<grid, block, 0, stream>>>(
        x, W1, b1, gamma, beta, W2, b2, W3, b3, out);
}